// GCNs_1185410973822
// MI455X (gfx1250) — compile-verified
//
#include <hip/hip_runtime.h>
#include <stdint.h>

// ---------------------------------------------------------------------------
// GCNII-style scalar-feature GCN, 49 layers, 500K nodes, 16M edges.
// Memory-bound SpMV; packed uint2 edge stream (128 MB) is L2-resident on
// MI455X (192 MB L2). Edge kernel streams via gfx1250 async global->LDS
// double buffering (ASYNCcnt) and uses HW f32 global atomics.
// ---------------------------------------------------------------------------

static __device__ __forceinline__ void atomicAddF(float* p, float v) {
  __hip_atomic_fetch_add(p, v, __ATOMIC_RELAXED, __HIP_MEMORY_SCOPE_AGENT);
}

__global__ void k_fill(float* __restrict__ deg, float* __restrict__ hh, int n) {
  int i = blockIdx.x * blockDim.x + threadIdx.x;
  if (i < n) { deg[i] = 1.0f; hh[i] = 0.0f; }   // deg starts at 1 (self loop)
}

// One pass over int64 adj: pack (src,dst) to uint2 and accumulate in-degree.
__global__ void k_degpack(const long long* __restrict__ src,
                          const long long* __restrict__ dst,
                          uint2* __restrict__ edges,
                          float* __restrict__ deg, int E) {
  int i = blockIdx.x * blockDim.x + threadIdx.x;
  if (i < E) {
    unsigned s = (unsigned)src[i];
    unsigned d = (unsigned)dst[i];
    edges[i] = make_uint2(s, d);
    atomicAddF(&deg[d], 1.0f);
  }
}

// Fallback (no packing): degree only.
__global__ void k_deg(const long long* __restrict__ dst,
                      float* __restrict__ deg, int E) {
  int i = blockIdx.x * blockDim.x + threadIdx.x;
  if (i < E) atomicAddF(&deg[(unsigned)dst[i]], 1.0f);
}

__global__ void k_dis(float* __restrict__ dis, const float* __restrict__ deg, int n) {
  int i = blockIdx.x * blockDim.x + threadIdx.x;
  if (i < n) dis[i] = rsqrtf(deg[i]);
}

__global__ void k_init(float* __restrict__ xs, const float* __restrict__ dis,
                       const float* __restrict__ x0, int n) {
  int i = blockIdx.x * blockDim.x + threadIdx.x;
  if (i < n) xs[i] = dis[i] * x0[i];
}

// ---------------------------------------------------------------------------
// Hot loop: hh[dst] += xs[src] over packed uint2 edges.
// Double-buffered async global->LDS staging: each wave's lanes stage their own
// 16B slot (2 edges) with global_load_async_to_lds_b128, overlap with atomics
// on the previous tile, synchronize with s_wait_asynccnt. Each lane reads only
// the LDS slot it loaded -> no workgroup barrier needed.
// ---------------------------------------------------------------------------
#define EDGE_BLOCK 256
#define TILE_EDGES 512   // EDGE_BLOCK lanes * 2 edges

__global__ __launch_bounds__(EDGE_BLOCK)
void k_edge_packed(const uint2* __restrict__ edges,
                   const float* __restrict__ xs,
                   float* __restrict__ hh, int E) {
  __shared__ alignas(16) uint4 buf[2][EDGE_BLOCK];
  const int tid    = threadIdx.x;
  const int ntiles = (E + TILE_EDGES - 1) / TILE_EDGES;
  const unsigned lds0 = (unsigned)(size_t)&buf[0][tid];
  const unsigned lds1 = (unsigned)(size_t)&buf[1][tid];

  int tile = blockIdx.x;
  const int stride = gridDim.x;
  if (tile >= ntiles) return;

  { // prime buffer 0
    int base = tile * TILE_EDGES + tid * 2;
    int idx  = (base <= E - 2) ? base : 0;   // clamp: safe, guarded at use
    const uint2* g = edges + idx;
    asm volatile("global_load_async_to_lds_b128 %0, %1, off"
                 :: "v"(lds0), "v"(g) : "memory");
  }

  int bufsel = 0;
  for (; tile < ntiles; tile += stride) {
    const int ntile = tile + stride;
    if (ntile < ntiles) {
      int base = ntile * TILE_EDGES + tid * 2;
      int idx  = (base <= E - 2) ? base : 0;
      const uint2* g = edges + idx;
      unsigned nlds = bufsel ? lds0 : lds1;
      asm volatile("global_load_async_to_lds_b128 %0, %1, off"
                   :: "v"(nlds), "v"(g) : "memory");
      asm volatile("s_wait_asynccnt 0x1" ::: "memory");  // current tile ready
    } else {
      asm volatile("s_wait_asynccnt 0x0" ::: "memory");  // last tile
    }

    uint4 e2 = buf[bufsel][tid];            // 2 packed edges from LDS
    int base = tile * TILE_EDGES + tid * 2; // E is even on the fast path
    if (base + 1 < E) {
      atomicAddF(&hh[e2.y], xs[e2.x]);
      atomicAddF(&hh[e2.w], xs[e2.z]);
    }
    bufsel ^= 1;
  }
}

// Fallback: direct int64 edge reads (no packing workspace available).
__global__ void k_edge_i64(const long long* __restrict__ src,
                           const long long* __restrict__ dst,
                           const float* __restrict__ xs,
                           float* __restrict__ hh, int E) {
  int i  = blockIdx.x * blockDim.x + threadIdx.x;
  int gs = gridDim.x * blockDim.x;
  for (; i < E; i += gs)
    atomicAddF(&hh[(unsigned)dst[i]], xs[(unsigned)src[i]]);
}

// Node update: h = dis*(hh + xs) (self loop folded in), combine with x0,
// scale by w[layer], activation, rescale to xs, reset hh for next layer.
// act: 0=identity(layer 0), 1=leaky_relu(0.01), 2=sigmoid, 3=relu,
//      4=final n*sigmoid -> int64 output.
__global__ void k_node(const float* __restrict__ dis, float* __restrict__ xs,
                       float* __restrict__ hh, const float* __restrict__ x0,
                       const float* __restrict__ w, int layer, int act,
                       float nf, long long* __restrict__ out, int n) {
  int i = blockIdx.x * blockDim.x + threadIdx.x;
  if (i >= n) return;
  const float wl = w[layer];
  const float di = dis[i];
  const float h  = di * (hh[i] + xs[i]);
  const float y  = (0.7f * h + 0.3f * x0[i]) * wl;
  float x;
  if (act == 0)      x = y;
  else if (act == 1) x = (y > 0.0f) ? y : 0.01f * y;
  else if (act == 2) x = 1.0f / (1.0f + expf(-y));
  else if (act == 4) x = nf * (1.0f / (1.0f + expf(-y)));
  else               x = fmaxf(y, 0.0f);
  hh[i] = 0.0f;                      // ready for next layer's accumulation
  if (act == 4) out[i] = (long long)x;   // astype(int64): truncation (x >= 0)
  else          xs[i]  = di * x;
}

extern "C" void kernel_launch(void* const* d_in, const int* in_sizes, int n_in,
                              void* d_out, int out_size, void* d_ws, size_t ws_size,
                              hipStream_t stream) {
  const float*     x0  = (const float*)d_in[0];
  const float*     w   = (const float*)d_in[1];
  const long long* adj = (const long long*)d_in[2];
  const int N  = in_sizes[0];
  const int NL = in_sizes[1];
  const int E  = in_sizes[2] / 2;
  const long long* src = adj;
  const long long* dst = adj + E;
  long long* out = (long long*)d_out;

  char* ws = (char*)d_ws;
  size_t off = 0;
  auto carve = [&](size_t bytes) -> void* {
    void* p = ws + off;
    off = (off + bytes + 255) & ~(size_t)255;
    return p;
  };
  float* deg = (float*)carve((size_t)N * 4);
  float* dis = (float*)carve((size_t)N * 4);
  float* xs  = (float*)carve((size_t)N * 4);
  float* hh  = (float*)carve((size_t)N * 4);
  uint2* edges = (uint2*)(ws + off);
  const bool fast = (off + (size_t)E * 8 <= ws_size) && (E % 2 == 0) && (E >= 2);

  const int TB = 256;
  const int gn = (N + TB - 1) / TB;
  const int ge = (E + TB - 1) / TB;

  k_fill<<<gn, TB, 0, stream>>>(deg, hh, N);
  if (fast) k_degpack<<<ge, TB, 0, stream>>>(src, dst, edges, deg, E);
  else      k_deg    <<<ge, TB, 0, stream>>>(dst, deg, E);
  k_dis <<<gn, TB, 0, stream>>>(dis, deg, N);
  k_init<<<gn, TB, 0, stream>>>(xs, dis, x0, N);

  for (int i = 0; i < NL; ++i) {
    int act;
    if (i == 0)            act = 0;
    else if (i == NL - 1)  act = 4;
    else if (i % 10 == 1)  act = 1;   // {1,11,21,31,41} leaky
    else if (i % 10 == 4)  act = 2;   // {4,14,24,34,44} sigmoid
    else                   act = 3;   // relu
    if (fast) {
      const int ntiles = (E + TILE_EDGES - 1) / TILE_EDGES;
      const int grid   = ntiles < 4096 ? ntiles : 4096;
      k_edge_packed<<<grid, EDGE_BLOCK, 0, stream>>>(edges, xs, hh, E);
    } else {
      k_edge_i64<<<8192, TB, 0, stream>>>(src, dst, xs, hh, E);
    }
    k_node<<<gn, TB, 0, stream>>>(dis, xs, hh, x0, w, i, act, (float)N, out, N);
  }
}